// FNS_88304527606498
// MI455X (gfx1250) — compile-verified
//
#include <hip/hip_runtime.h>

// ---------------------------------------------------------------------------
// Sizes (fixed by the reference): B=16, C=1, N1=511, CH=8, N2=1024, H=513
// ---------------------------------------------------------------------------
#define NB   16
#define N1D  511
#define HD   513
#define CHN  8
static const long HH_ = 513l * 513;   // 263169  (H x H)
static const long HW_ = 513l * 511;   // 262143  (H x N1 and N1 x H)
static const long NN_ = 511l * 511;   // 261121  (N1 x N1)

typedef float v2f __attribute__((ext_vector_type(2)));
typedef float v8f __attribute__((ext_vector_type(8)));

__device__ __forceinline__ float2 cmadd(float2 acc, float2 a, float2 b) {
    acc.x += a.x * b.x - a.y * b.y;
    acc.y += a.x * b.y + a.y * b.x;
    return acc;
}

// Low 32 bits of a flat shared-memory pointer == LDS byte address (ISA 10.2).
__device__ __forceinline__ unsigned lds_addr_of(const void* p) {
    return (unsigned)(unsigned long long)(uintptr_t)p;
}

// ---------------------------------------------------------------------------
// Basis matrices for the sine/Fourier transforms (device-generated, exact
// periodic reduction: argument = 2*pi/1024 * ((..) mod 1024)).
//   S  [513 x 511] : S[p,m]  = sin(2*pi*(p-256)*(m+1)/1024)
//   St [511 x 513] : S^T
//   Am [511 x 513] : cos(2*pi*m*(p-255)/1024)
//   Bm [511 x 513] : -sin(2*pi*m*(p-255)/1024)      (F = Am + i*Bm)
//   At [513 x 511], Bt [513 x 511] : transposes
// ---------------------------------------------------------------------------
__global__ void build_basis_kernel(float* __restrict__ S,  float* __restrict__ St,
                                   float* __restrict__ Am, float* __restrict__ Bm,
                                   float* __restrict__ At, float* __restrict__ Bt) {
    int idx = blockIdx.x * blockDim.x + threadIdx.x;
    if (idx >= HD * HD) return;
    int p = idx / HD, q = idx % HD;
    const float w0 = 6.283185307179586476f / 1024.0f;
    if (q < N1D) {
        long t = (long)(p - 256) * (long)(q + 1);
        float s = sinf(w0 * (float)(int)(t % 1024));
        S[(long)p * N1D + q]  = s;
        St[(long)q * HD + p]  = s;
    }
    if (p < N1D) {
        long t = (long)p * (long)(q - 255);
        float ang = w0 * (float)(int)(t % 1024);
        float cv = cosf(ang), sv = sinf(ang);
        Am[(long)p * HD + q]  = cv;
        Bm[(long)p * HD + q]  = -sv;
        At[(long)q * N1D + p] = cv;
        Bt[(long)q * N1D + p] = -sv;
    }
}

// ---------------------------------------------------------------------------
// f32 WMMA GEMM:  C = alpha * A(MxK) * B(KxN)  [+ C if accum]
// Block tile 32x64, 256 threads = 8 wave32 waves (2x4 wave grid), each wave
// producing one 16x16 tile via V_WMMA_F32_16X16X4_F32.
// Double-buffered LDS staging: interior slabs use GLOBAL_LOAD_ASYNC_TO_LDS_B128
// (ASYNCcnt) issued for slab k+1 while slab k feeds the WMMAs; edge slabs use
// a guarded zero-fill VALU path. Batched with blockIdx.z element strides.
// Per-lane layouts follow the CDNA5 ISA:
//   A 16x4 : lane l holds row M=l%16, elems K = 2*(l/16)+{0,1}
//   B 4x16 : lane l holds col N=l%16, elems K = 2*(l/16)+{0,1}
//   C/D    : lane l holds col N=l%16, VGPR j -> row M = j + 8*(l/16)
// ---------------------------------------------------------------------------
__global__ __launch_bounds__(256) void gemm_wmma_kernel(
    const float* __restrict__ A, const float* __restrict__ B, float* __restrict__ C,
    int M, int N, int K, int lda, int ldb, int ldc,
    long sA, long sB, long sC, float alpha, int accum) {
    A += (long)blockIdx.z * sA;
    B += (long)blockIdx.z * sB;
    C += (long)blockIdx.z * sC;

    __shared__ float As[2][32][33];
    __shared__ float Bs[2][32][65];

    const int tid  = threadIdx.x;
    const int lane = tid & 31;
    const int wave = tid >> 5;
    const int wm   = (wave >> 2) << 4;   // 0 / 16 : wave row offset in tile
    const int wn   = (wave & 3) << 4;    // 0..48  : wave col offset in tile
    const int g    = lane >> 4;          // lane half
    const int lr   = lane & 15;
    const int bm   = blockIdx.x * 32;
    const int bn   = blockIdx.y * 64;

    const bool mnFull = (bm + 32 <= M) && (bn + 64 <= N);

    // Stage one K-slab into LDS buffer `buf`.
    auto stage = [&](int kc, int buf) {
        if (mnFull && (kc + 32 <= K)) {
            // Fully interior: async 128-bit global -> LDS copies (no VGPR data).
            {
                int r  = tid >> 3;            // 0..31
                int c4 = (tid & 7) << 2;      // 0..28
                unsigned ldsa = lds_addr_of(&As[buf][r][c4]);
                unsigned long long ga =
                    (unsigned long long)(uintptr_t)(A + (long)(bm + r) * lda + (kc + c4));
                asm volatile("global_load_async_to_lds_b128 %0, %1, off"
                             :: "v"(ldsa), "v"(ga) : "memory");
            }
            {
                int r  = tid >> 4;            // 0..15
                int c4 = (tid & 15) << 2;     // 0..60
                unsigned ldsb0 = lds_addr_of(&Bs[buf][r][c4]);
                unsigned ldsb1 = lds_addr_of(&Bs[buf][r + 16][c4]);
                unsigned long long gb0 =
                    (unsigned long long)(uintptr_t)(B + (long)(kc + r) * ldb + (bn + c4));
                unsigned long long gb1 =
                    (unsigned long long)(uintptr_t)(B + (long)(kc + r + 16) * ldb + (bn + c4));
                asm volatile("global_load_async_to_lds_b128 %0, %1, off"
                             :: "v"(ldsb0), "v"(gb0) : "memory");
                asm volatile("global_load_async_to_lds_b128 %0, %1, off"
                             :: "v"(ldsb1), "v"(gb1) : "memory");
            }
        } else {
            // Edge slab: guarded, zero-filled VALU path.
            for (int i = tid; i < 32 * 32; i += 256) {
                int r = i >> 5, c = i & 31;
                int gr = bm + r, gk = kc + c;
                As[buf][r][c] = (gr < M && gk < K) ? A[(long)gr * lda + gk] : 0.0f;
            }
            for (int i = tid; i < 32 * 64; i += 256) {
                int r = i >> 6, c = i & 63;
                int gk = kc + r, gc = bn + c;
                Bs[buf][r][c] = (gk < K && gc < N) ? B[(long)gk * ldb + gc] : 0.0f;
            }
        }
    };

    v8f acc = {};
    int buf = 0;
    stage(0, 0);

    for (int kc = 0; kc < K; kc += 32) {
        // Make this wave's async copies visible, then rendezvous.
        asm volatile("s_wait_asynccnt 0x0" ::: "memory");
        __syncthreads();

        // Prefetch next slab into the other buffer while we compute.
        if (kc + 32 < K) stage(kc + 32, buf ^ 1);

#pragma unroll
        for (int kk = 0; kk < 32; kk += 4) {
            int ks = kk + 2 * g;
            v2f a, b;
            a[0] = As[buf][wm + lr][ks];
            a[1] = As[buf][wm + lr][ks + 1];
            b[0] = Bs[buf][ks][wn + lr];
            b[1] = Bs[buf][ks + 1][wn + lr];
            acc = __builtin_amdgcn_wmma_f32_16x16x4_f32(
                false, a, false, b, (short)0, acc, false, false);
        }
        __syncthreads();   // all reads of `buf` done before it is restaged
        buf ^= 1;
    }

#pragma unroll
    for (int j = 0; j < 8; j++) {
        int row = bm + wm + 8 * g + j;
        int col = bn + wn + lr;
        if (row < M && col < N) {
            float v = alpha * acc[j];
            if (accum) v += C[(long)row * ldc + col];
            C[(long)row * ldc + col] = v;
        }
    }
}

static inline void launch_gemm(const float* A, const float* B, float* C,
                               int M, int N, int K, int lda, int ldb, int ldc,
                               long sA, long sB, long sC,
                               float alpha, int accum, int batch, hipStream_t s) {
    dim3 grid((M + 31) / 32, (N + 63) / 64, batch);
    gemm_wmma_kernel<<<grid, 256, 0, s>>>(A, B, C, M, N, K, lda, ldb, ldc,
                                          sA, sB, sC, alpha, accum);
}

// ---------------------------------------------------------------------------
// Complex 3x3 convs (cross-correlation, pad=1) on 513x513 images.
// ---------------------------------------------------------------------------

// conv1 forward: real input (r_hat slice) -> 8 complex channels
__global__ void conv_first_kernel(const float* __restrict__ in,
                                  const float2* __restrict__ w,   // [8][1][3][3]
                                  float2* __restrict__ out) {
    __shared__ float  tile[18][18];
    __shared__ float2 wsh[72];
    int tx = threadIdx.x, ty = threadIdx.y, tid = ty * 16 + tx;
    int bx = blockIdx.x * 16, by = blockIdx.y * 16;
    if (tid < 72) wsh[tid] = w[tid];
    for (int i = tid; i < 18 * 18; i += 256) {
        int r = i / 18, c = i % 18;
        int gy = by + r - 1, gx = bx + c - 1;
        tile[r][c] = (gy >= 0 && gy < HD && gx >= 0 && gx < HD)
                         ? in[(long)gy * HD + gx] : 0.0f;
    }
    __syncthreads();
    int x = bx + tx, y = by + ty;
    if (x >= HD || y >= HD) return;
    float v[9];
#pragma unroll
    for (int t = 0; t < 9; t++) v[t] = tile[ty + t / 3][tx + t % 3];
#pragma unroll
    for (int co = 0; co < CHN; co++) {
        float2 acc = make_float2(0.f, 0.f);
#pragma unroll
        for (int t = 0; t < 9; t++) {
            float2 wv = wsh[co * 9 + t];
            acc.x += wv.x * v[t];
            acc.y += wv.y * v[t];
        }
        out[(long)co * HH_ + (long)y * HD + x] = acc;
    }
}

// 8ch -> 8ch complex conv. mode==0: w[co][ci][ky][kx];
// mode==1 (conj-transpose): conj(w[ci][co][kx][ky]) per the reference wt.
__global__ void conv8_kernel(const float2* __restrict__ in,
                             const float2* __restrict__ w,    // [8][8][3][3]
                             float2* __restrict__ out, int mode) {
    __shared__ float2 tile[CHN][18][18];  // ~20.7 KB
    __shared__ float2 wsh[CHN][CHN][9];   //  ~4.6 KB
    int tx = threadIdx.x, ty = threadIdx.y, tid = ty * 16 + tx;
    int bx = blockIdx.x * 16, by = blockIdx.y * 16;
    for (int i = tid; i < CHN * CHN * 9; i += 256) {
        int co = i / 72, rem = i % 72, ci = rem / 9, t = rem % 9;
        int ky = t / 3, kx = t % 3;
        float2 wv;
        if (mode == 0) {
            wv = w[(co * CHN + ci) * 9 + ky * 3 + kx];
        } else {
            wv = w[(ci * CHN + co) * 9 + kx * 3 + ky];
            wv.y = -wv.y;
        }
        wsh[co][ci][t] = wv;
    }
    for (int i = tid; i < CHN * 18 * 18; i += 256) {
        int ch = i / 324, rem = i % 324, r = rem / 18, c = rem % 18;
        int gy = by + r - 1, gx = bx + c - 1;
        tile[ch][r][c] = (gy >= 0 && gy < HD && gx >= 0 && gx < HD)
                             ? in[(long)ch * HH_ + (long)gy * HD + gx]
                             : make_float2(0.f, 0.f);
    }
    __syncthreads();
    int x = bx + tx, y = by + ty;
    if (x >= HD || y >= HD) return;
    float2 acc[CHN];
#pragma unroll
    for (int co = 0; co < CHN; co++) acc[co] = make_float2(0.f, 0.f);
    for (int ci = 0; ci < CHN; ci++) {
        float2 v[9];
#pragma unroll
        for (int t = 0; t < 9; t++) v[t] = tile[ci][ty + t / 3][tx + t % 3];
#pragma unroll
        for (int co = 0; co < CHN; co++) {
#pragma unroll
            for (int t = 0; t < 9; t++) acc[co] = cmadd(acc[co], wsh[co][ci][t], v[t]);
        }
    }
#pragma unroll
    for (int co = 0; co < CHN; co++)
        out[(long)co * HH_ + (long)y * HD + x] = acc[co];
}

// Last backward conv: 8 complex channels -> 1, weights = conj-transposed w1.
// Writes planar real/imag for the inverse-transform GEMMs.
__global__ void conv_last_kernel(const float2* __restrict__ in,
                                 const float2* __restrict__ w1,  // [8][1][3][3]
                                 float* __restrict__ Xr, float* __restrict__ Xi) {
    __shared__ float2 tile[CHN][18][18];
    __shared__ float2 wsh[72];
    int tx = threadIdx.x, ty = threadIdx.y, tid = ty * 16 + tx;
    int bx = blockIdx.x * 16, by = blockIdx.y * 16;
    if (tid < 72) {
        int ci = tid / 9, t = tid % 9, ky = t / 3, kx = t % 3;
        float2 wv = w1[ci * 9 + kx * 3 + ky];  // spatial transpose
        wv.y = -wv.y;                          // conjugate
        wsh[tid] = wv;
    }
    for (int i = tid; i < CHN * 18 * 18; i += 256) {
        int ch = i / 324, rem = i % 324, r = rem / 18, c = rem % 18;
        int gy = by + r - 1, gx = bx + c - 1;
        tile[ch][r][c] = (gy >= 0 && gy < HD && gx >= 0 && gx < HD)
                             ? in[(long)ch * HH_ + (long)gy * HD + gx]
                             : make_float2(0.f, 0.f);
    }
    __syncthreads();
    int x = bx + tx, y = by + ty;
    if (x >= HD || y >= HD) return;
    float2 acc = make_float2(0.f, 0.f);
    for (int ci = 0; ci < CHN; ci++) {
#pragma unroll
        for (int t = 0; t < 9; t++)
            acc = cmadd(acc, wsh[ci * 9 + t], tile[ci][ty + t / 3][tx + t % 3]);
    }
    Xr[(long)y * HD + x] = acc.x;
    Xi[(long)y * HD + x] = acc.y;
}

// out_ft = r_hat * theta  (theta broadcast over the 8 channels)
__global__ void theta_mul_kernel(float2* __restrict__ a,
                                 const float2* __restrict__ th) {
    long idx = (long)blockIdx.x * blockDim.x + threadIdx.x;
    if (idx >= (long)CHN * HH_) return;
    long p = idx % HH_;
    float2 x = a[idx], t = th[p];
    a[idx] = make_float2(x.x * t.x - x.y * t.y, x.x * t.y + x.y * t.x);
}

// ---------------------------------------------------------------------------
// Orchestration
// ---------------------------------------------------------------------------
extern "C" void kernel_launch(void* const* d_in, const int* in_sizes, int n_in,
                              void* d_out, int out_size, void* d_ws, size_t ws_size,
                              hipStream_t stream) {
    (void)in_sizes; (void)n_in; (void)out_size; (void)ws_size;

    const float*  r     = (const float*)d_in[0];
    const float2* w1    = (const float2*)d_in[1];   // [16][8][1][3][3]
    const float2* w2    = (const float2*)d_in[2];   // [16][8][8][3][3]
    const float2* w3    = (const float2*)d_in[3];   // [16][8][8][3][3]
    const float2* theta = (const float2*)d_in[4];   // [16][1][513][513]
    float* out = (float*)d_out;                     // [16][1][511][511]

    // Workspace carve-up (floats)
    float* ws  = (float*)d_ws;
    float* S   = ws;  ws += HW_;
    float* St  = ws;  ws += HW_;
    float* Am  = ws;  ws += HW_;
    float* Bm  = ws;  ws += HW_;
    float* At  = ws;  ws += HW_;
    float* Bt  = ws;  ws += HW_;
    float* T1   = ws; ws += (long)NB * HW_;   // S * r            [16][513][511]
    float* Rhat = ws; ws += (long)NB * HH_;   // sine transform   [16][513][513]
    float* Xr   = ws; ws += (long)NB * HH_;   // conv-chain out   [16][513][513]
    float* Xi   = ws; ws += (long)NB * HH_;
    float* M1   = ws; ws += (long)NB * HW_;   // [16][511][513]
    float* M2   = ws; ws += (long)NB * HW_;
    float2* cA  = (float2*)ws; ws += 2l * CHN * HH_;   // ping
    float2* cB  = (float2*)ws; ws += 2l * CHN * HH_;   // pong

    // 1) Basis matrices
    {
        long n = (long)HD * HD;
        build_basis_kernel<<<(n + 255) / 256, 256, 0, stream>>>(S, St, Am, Bm, At, Bt);
    }

    // 2) Forward sine transform, batched over z:
    //    T1 = S * r ; Rhat = c * T1 * S^T,  c = -4/1024^2
    const float cscale = -4.0f / (1024.0f * 1024.0f);
    launch_gemm(S,  r,  T1,   HD,  N1D, N1D, N1D, N1D, N1D, 0,   NN_, HW_, 1.0f,  0, NB, stream);
    launch_gemm(T1, St, Rhat, HD,  HD,  N1D, N1D, HD,  HD,  HW_, 0,   HH_, cscale,0, NB, stream);

    // 3) Per-batch complex conv chain (ping-pong, keeps workspace ~34 MB)
    dim3 cgrid((HD + 15) / 16, (HD + 15) / 16);
    dim3 cblk(16, 16);
    long thn = (long)CHN * HH_;
    for (int b = 0; b < NB; b++) {
        const float2* w1b = w1 + (long)b * 72;
        const float2* w2b = w2 + (long)b * 576;
        const float2* w3b = w3 + (long)b * 576;
        conv_first_kernel<<<cgrid, cblk, 0, stream>>>(Rhat + (long)b * HH_, w1b, cA);
        conv8_kernel<<<cgrid, cblk, 0, stream>>>(cA, w2b, cB, 0);
        conv8_kernel<<<cgrid, cblk, 0, stream>>>(cB, w3b, cA, 0);
        theta_mul_kernel<<<(thn + 255) / 256, 256, 0, stream>>>(cA, theta + (long)b * HH_);
        conv8_kernel<<<cgrid, cblk, 0, stream>>>(cA, w3b, cB, 1);
        conv8_kernel<<<cgrid, cblk, 0, stream>>>(cB, w2b, cA, 1);
        conv_last_kernel<<<cgrid, cblk, 0, stream>>>(cA, w1b,
                                                     Xr + (long)b * HH_,
                                                     Xi + (long)b * HH_);
    }

    // 4) Inverse transform, batched over z:
    //    M1 = Am*Xr - Bm*Xi ; M2 = Am*Xi + Bm*Xr ; e = M1*At - M2*Bt
    launch_gemm(Am, Xr, M1, N1D, HD, HD, HD, HD, HD, 0, HH_, HW_,  1.0f, 0, NB, stream);
    launch_gemm(Bm, Xi, M1, N1D, HD, HD, HD, HD, HD, 0, HH_, HW_, -1.0f, 1, NB, stream);
    launch_gemm(Am, Xi, M2, N1D, HD, HD, HD, HD, HD, 0, HH_, HW_,  1.0f, 0, NB, stream);
    launch_gemm(Bm, Xr, M2, N1D, HD, HD, HD, HD, HD, 0, HH_, HW_,  1.0f, 1, NB, stream);
    launch_gemm(M1, At, out, N1D, N1D, HD, HD, N1D, N1D, HW_, 0, NN_,  1.0f, 0, NB, stream);
    launch_gemm(M2, Bt, out, N1D, N1D, HD, HD, N1D, N1D, HW_, 0, NN_, -1.0f, 1, NB, stream);
}